// NetworkFlowGCN_40913858462143
// MI455X (gfx1250) — compile-verified
//
#include <hip/hip_runtime.h>
#include <hip/hip_bf16.h>

#define IN_DIM 64
#define HID    128
#define C2D    64
#define C3D    32
#define HEADS  4
#define NCLS   5
#define NGRAPH 64

typedef float v2f __attribute__((ext_vector_type(2)));
typedef float v8f __attribute__((ext_vector_type(8)));

__device__ __forceinline__ void atomicAddF(float* p, float v) {
  unsafeAtomicAdd(p, v);   // global_atomic_add_f32
}

// float atomic max via signed/unsigned integer ordering trick
__device__ __forceinline__ void atomicMaxF(float* addr, float value) {
  if (value >= 0.0f) atomicMax((int*)addr, __float_as_int(value));
  else               atomicMin((unsigned int*)addr, __float_as_uint(value));
}

__device__ __forceinline__ float leaky02(float x) { return x > 0.0f ? x : 0.2f * x; }

// src/dst for extended edge list (E real edges + N self loops)
__device__ __forceinline__ void edge_sd(const int* ei, int E, int idx, int& s, int& d) {
  if (idx < E) { s = ei[idx]; d = ei[E + idx]; }
  else         { s = idx - E; d = s; }
}

// ---------------------------------------------------------------------------
// Dense GEMM  C[nrows x NC] = A[nrows x K] @ W[K x NC]  using fp32 WMMA.
// One wave32 owns a 16-row strip and all NC/16 column tiles.
// A frag (16x4 f32): vgpr0 = K = kh,   vgpr1 = K = kh+1   (kh = 0 | 2 by half-wave)
// B frag (4x16 f32): vgpr0 = row kh,   vgpr1 = row kh+1
// C/D (16x16 f32):   vgpr j -> M = 8*(lane>=16) + j, N = lane%16
// ---------------------------------------------------------------------------
template<int K, int NC>
__global__ __launch_bounds__(256) void gemm_wmma_f32(const float* __restrict__ A,
                                                     const float* __restrict__ W,
                                                     float* __restrict__ C, int nrows) {
  const int wave = blockIdx.x * (blockDim.x >> 5) + (threadIdx.x >> 5);
  const int lane = threadIdx.x & 31;
  const int row0 = wave * 16;
  if (row0 >= nrows) return;                  // wave-uniform: EXEC stays all-ones
  constexpr int NT = NC / 16;
  v8f acc[NT];
#pragma unroll
  for (int t = 0; t < NT; ++t) acc[t] = {};

  const int lm = lane & 15;
  const int kh = (lane >> 4) << 1;            // 0 or 2
  const int ar = (row0 + lm < nrows) ? (row0 + lm) : (nrows - 1);
  const float* aptr = A + (size_t)ar * K + kh;

  for (int k = 0; k < K; k += 4) {
    v2f af;
    af.x = aptr[k];
    af.y = aptr[k + 1];
    const float* wptr = W + (size_t)(k + kh) * NC + lm;
#pragma unroll
    for (int t = 0; t < NT; ++t) {
      v2f bf;
      bf.x = wptr[t * 16];
      bf.y = wptr[t * 16 + NC];
      acc[t] = __builtin_amdgcn_wmma_f32_16x16x4_f32(false, af, false, bf,
                                                     (short)0, acc[t], false, false);
    }
  }

  const int mb = row0 + ((lane >> 4) << 3);
#pragma unroll
  for (int t = 0; t < NT; ++t)
#pragma unroll
    for (int j = 0; j < 8; ++j) {
      int r = mb + j;
      if (r < nrows) C[(size_t)r * NC + t * 16 + lm] = acc[t][j];
    }
}

// ---------------------------------------------------------------------------
__global__ void fill_kernel(float* __restrict__ p, float v, long n) {
  long i = (long)blockIdx.x * blockDim.x + threadIdx.x;
  long stride = (long)gridDim.x * blockDim.x;
  for (; i < n; i += stride) p[i] = v;
}

__global__ void deg_kernel(const int* __restrict__ ei, float* __restrict__ deg, int E) {
  int e = blockIdx.x * blockDim.x + threadIdx.x;
  if (e < E) atomicAddF(&deg[ei[E + e]], 1.0f);
}

__global__ void rsqrt_kernel(float* __restrict__ d, int n) {
  int i = blockIdx.x * blockDim.x + threadIdx.x;
  if (i < n) d[i] = rsqrtf(d[i]);
}

// one wave per edge: coalesced gather of lin[src][:], atomic scatter to agg[dst][:]
template<int D>
__global__ __launch_bounds__(256) void gcn_agg_kernel(const float* __restrict__ lin,
                                                      const int* __restrict__ ei,
                                                      const float* __restrict__ dinv,
                                                      float* __restrict__ agg, int E) {
  int wave = blockIdx.x * (blockDim.x >> 5) + (threadIdx.x >> 5);
  int lane = threadIdx.x & 31;
  if (wave >= E) return;
  int s = ei[wave], d = ei[E + wave];
  float nrm = dinv[s] * dinv[d];
  const float* hs = lin + (size_t)s * D;
  float* od = agg + (size_t)d * D;
#pragma unroll
  for (int f = lane; f < D; f += 32) atomicAddF(od + f, nrm * hs[f]);
}

// self-loop term + bias + BatchNorm + ReLU
template<int D>
__global__ void gcn_post_kernel(const float* __restrict__ lin, const float* __restrict__ agg,
                                const float* __restrict__ dinv, const float* __restrict__ b,
                                const float* __restrict__ g, const float* __restrict__ be,
                                const float* __restrict__ m, const float* __restrict__ v,
                                float* __restrict__ out, int n) {
  long t = (long)blockIdx.x * blockDim.x + threadIdx.x;
  if (t >= (long)n * D) return;
  int i = (int)(t / D), f = (int)(t % D);
  float di = dinv[i];
  float val = agg[t] + di * di * lin[t] + b[f];
  val = (val - m[f]) * rsqrtf(v[f] + 1e-5f) * g[f] + be[f];
  out[t] = fmaxf(val, 0.0f);
}

// per-node, per-head attention coefficients a_s, a_d
__global__ void gat_attn_kernel(const float* __restrict__ hh, const float* __restrict__ ags,
                                const float* __restrict__ agd, float* __restrict__ a_s,
                                float* __restrict__ a_d, int n) {
  int t = blockIdx.x * blockDim.x + threadIdx.x;
  if (t >= n * HEADS) return;
  int i = t >> 2, h = t & 3;
  const float* hv = hh + (size_t)i * (HEADS * C3D) + h * C3D;
  float ss = 0.0f, sd = 0.0f;
#pragma unroll
  for (int d2 = 0; d2 < C3D; ++d2) {
    ss += hv[d2] * ags[h * C3D + d2];
    sd += hv[d2] * agd[h * C3D + d2];
  }
  a_s[t] = ss;
  a_d[t] = sd;
}

__global__ void gat_max_kernel(const int* __restrict__ ei, const float* __restrict__ a_s,
                               const float* __restrict__ a_d, float* __restrict__ emax,
                               int E, int total) {
  int t = blockIdx.x * blockDim.x + threadIdx.x;
  if (t >= total * HEADS) return;
  int idx = t >> 2, h = t & 3;
  int s, d; edge_sd(ei, E, idx, s, d);
  float e = leaky02(a_s[s * HEADS + h] + a_d[d * HEADS + h]);
  atomicMaxF(&emax[d * HEADS + h], e);
}

__global__ void gat_den_kernel(const int* __restrict__ ei, const float* __restrict__ a_s,
                               const float* __restrict__ a_d, const float* __restrict__ emax,
                               float* __restrict__ den, int E, int total) {
  int t = blockIdx.x * blockDim.x + threadIdx.x;
  if (t >= total * HEADS) return;
  int idx = t >> 2, h = t & 3;
  int s, d; edge_sd(ei, E, idx, s, d);
  float e = leaky02(a_s[s * HEADS + h] + a_d[d * HEADS + h]);
  atomicAddF(&den[d * HEADS + h], expf(e - emax[d * HEADS + h]));
}

// one wave per extended edge; lane = feature (C3D == 32); fuse mean over heads
__global__ __launch_bounds__(256) void gat_agg_kernel(const int* __restrict__ ei,
    const float* __restrict__ hh, const float* __restrict__ a_s, const float* __restrict__ a_d,
    const float* __restrict__ emax, const float* __restrict__ den,
    float* __restrict__ agg, int E, int total) {
  int wave = blockIdx.x * (blockDim.x >> 5) + (threadIdx.x >> 5);
  int lane = threadIdx.x & 31;
  if (wave >= total) return;
  int s, d; edge_sd(ei, E, wave, s, d);
  float val = 0.0f;
#pragma unroll
  for (int h = 0; h < HEADS; ++h) {
    float e = leaky02(a_s[s * HEADS + h] + a_d[d * HEADS + h]);
    float att = expf(e - emax[d * HEADS + h]) / den[d * HEADS + h];
    val += att * hh[(size_t)s * (HEADS * C3D) + h * C3D + lane];
  }
  atomicAddF(&agg[(size_t)d * C3D + lane], val);
}

__global__ void gat_post_kernel(const float* __restrict__ agg, const float* __restrict__ bg,
                                float* __restrict__ out, int n) {
  long t = (long)blockIdx.x * blockDim.x + threadIdx.x;
  if (t >= (long)n * C3D) return;
  int f = (int)(t % C3D);
  out[t] = fmaxf(agg[t] * (1.0f / HEADS) + bg[f], 0.0f);
}

__global__ __launch_bounds__(256) void pool_kernel(const float* __restrict__ h,
    const int* __restrict__ batch, float* __restrict__ pooled, float* __restrict__ cnt, int n) {
  int wave = blockIdx.x * (blockDim.x >> 5) + (threadIdx.x >> 5);
  int lane = threadIdx.x & 31;
  if (wave >= n) return;
  int b = batch[wave];
  atomicAddF(&pooled[b * C3D + lane], h[(size_t)wave * C3D + lane]);
  if (lane == 0) atomicAddF(&cnt[b], 1.0f);
}

__global__ void classifier_kernel(const float* __restrict__ pooled, const float* __restrict__ cnt,
                                  const float* __restrict__ Wc1, const float* __restrict__ bc1,
                                  const float* __restrict__ Wc2, const float* __restrict__ bc2,
                                  float* __restrict__ out) {
  int g = threadIdx.x;
  if (g >= NGRAPH) return;
  float inv = 1.0f / fmaxf(cnt[g], 1.0f);
  float p[C3D];
#pragma unroll
  for (int j = 0; j < C3D; ++j) p[j] = pooled[g * C3D + j] * inv;
  float o[NCLS];
#pragma unroll
  for (int c = 0; c < NCLS; ++c) o[c] = bc2[c];
  for (int k = 0; k < 16; ++k) {
    float hdn = bc1[k];
#pragma unroll
    for (int j = 0; j < C3D; ++j) hdn += p[j] * Wc1[j * 16 + k];
    hdn = fmaxf(hdn, 0.0f);
#pragma unroll
    for (int c = 0; c < NCLS; ++c) o[c] += hdn * Wc2[k * NCLS + c];
  }
#pragma unroll
  for (int c = 0; c < NCLS; ++c) out[g * NCLS + c] = o[c];
}

// ---------------------------------------------------------------------------
static inline int cdiv(long a, long b) { return (int)((a + b - 1) / b); }

extern "C" void kernel_launch(void* const* d_in, const int* in_sizes, int n_in,
                              void* d_out, int out_size, void* d_ws, size_t ws_size,
                              hipStream_t stream) {
  const float* x    = (const float*)d_in[0];
  const int*   ei   = (const int*)d_in[1];
  const int*   batch= (const int*)d_in[2];
  const float* W1 = (const float*)d_in[3];  const float* b1 = (const float*)d_in[4];
  const float* W2 = (const float*)d_in[5];  const float* b2 = (const float*)d_in[6];
  const float* W3 = (const float*)d_in[7];  const float* b3 = (const float*)d_in[8];
  const float* g1 = (const float*)d_in[9];  const float* be1 = (const float*)d_in[10];
  const float* m1 = (const float*)d_in[11]; const float* v1 = (const float*)d_in[12];
  const float* g2 = (const float*)d_in[13]; const float* be2 = (const float*)d_in[14];
  const float* m2 = (const float*)d_in[15]; const float* v2 = (const float*)d_in[16];
  const float* g3 = (const float*)d_in[17]; const float* be3 = (const float*)d_in[18];
  const float* m3 = (const float*)d_in[19]; const float* v3 = (const float*)d_in[20];
  const float* Wg  = (const float*)d_in[21];
  const float* ags = (const float*)d_in[22];
  const float* agd = (const float*)d_in[23];
  const float* bg  = (const float*)d_in[24];
  const float* Wc1 = (const float*)d_in[25]; const float* bc1 = (const float*)d_in[26];
  const float* Wc2 = (const float*)d_in[27]; const float* bc2 = (const float*)d_in[28];
  float* out = (float*)d_out;

  const int N = in_sizes[0] / IN_DIM;
  const int E = in_sizes[1] / 2;
  const int total = E + N;

  float* B1   = (float*)d_ws;                 // N x 128 (lin / hh)
  float* B2   = B1 + (size_t)N * HID;         // N x 128 (agg)
  float* B3   = B2 + (size_t)N * HID;         // N x 128 (activations)
  float* dinv = B3 + (size_t)N * HID;         // N  (deg -> d^-1/2 in place)
  float* a_s  = dinv + N;                     // N x 4
  float* a_d  = a_s + (size_t)N * HEADS;      // N x 4
  float* emax = a_d + (size_t)N * HEADS;      // N x 4
  float* den  = emax + (size_t)N * HEADS;     // N x 4
  float* pooled = den + (size_t)N * HEADS;    // 64 x 32
  float* cnt  = pooled + NGRAPH * C3D;        // 64

  dim3 tpb(256);
  const int gemm_blocks = cdiv(cdiv(N, 16), 8);

  // --- symmetric-normalization degrees (self loop folded into init=1) ---
  fill_kernel<<<cdiv(N, 256), tpb, 0, stream>>>(dinv, 1.0f, N);
  deg_kernel<<<cdiv(E, 256), tpb, 0, stream>>>(ei, dinv, E);
  rsqrt_kernel<<<cdiv(N, 256), tpb, 0, stream>>>(dinv, N);

  // --- GCN layer 1: 64 -> 128 ---
  gemm_wmma_f32<IN_DIM, HID><<<gemm_blocks, tpb, 0, stream>>>(x, W1, B1, N);
  fill_kernel<<<32768, tpb, 0, stream>>>(B2, 0.0f, (long)N * HID);
  gcn_agg_kernel<HID><<<cdiv(E, 8), tpb, 0, stream>>>(B1, ei, dinv, B2, E);
  gcn_post_kernel<HID><<<cdiv((long)N * HID, 256), tpb, 0, stream>>>(
      B1, B2, dinv, b1, g1, be1, m1, v1, B3, N);

  // --- GCN layer 2: 128 -> 64 ---
  gemm_wmma_f32<HID, C2D><<<gemm_blocks, tpb, 0, stream>>>(B3, W2, B1, N);
  fill_kernel<<<32768, tpb, 0, stream>>>(B2, 0.0f, (long)N * C2D);
  gcn_agg_kernel<C2D><<<cdiv(E, 8), tpb, 0, stream>>>(B1, ei, dinv, B2, E);
  gcn_post_kernel<C2D><<<cdiv((long)N * C2D, 256), tpb, 0, stream>>>(
      B1, B2, dinv, b2, g2, be2, m2, v2, B3, N);

  // --- GCN layer 3: 64 -> 32 ---
  gemm_wmma_f32<C2D, C3D><<<gemm_blocks, tpb, 0, stream>>>(B3, W3, B1, N);
  fill_kernel<<<16384, tpb, 0, stream>>>(B2, 0.0f, (long)N * C3D);
  gcn_agg_kernel<C3D><<<cdiv(E, 8), tpb, 0, stream>>>(B1, ei, dinv, B2, E);
  gcn_post_kernel<C3D><<<cdiv((long)N * C3D, 256), tpb, 0, stream>>>(
      B1, B2, dinv, b3, g3, be3, m3, v3, B3, N);

  // --- GAT: hh = h @ Wg (32 -> 4*32), attention softmax over incoming edges ---
  gemm_wmma_f32<C3D, HEADS * C3D><<<gemm_blocks, tpb, 0, stream>>>(B3, Wg, B1, N);
  gat_attn_kernel<<<cdiv((long)N * HEADS, 256), tpb, 0, stream>>>(B1, ags, agd, a_s, a_d, N);
  fill_kernel<<<cdiv((long)N * HEADS, 256), tpb, 0, stream>>>(emax, -3.402823e38f, (long)N * HEADS);
  gat_max_kernel<<<cdiv((long)total * HEADS, 256), tpb, 0, stream>>>(ei, a_s, a_d, emax, E, total);
  fill_kernel<<<cdiv((long)N * HEADS, 256), tpb, 0, stream>>>(den, 0.0f, (long)N * HEADS);
  gat_den_kernel<<<cdiv((long)total * HEADS, 256), tpb, 0, stream>>>(ei, a_s, a_d, emax, den, E, total);
  fill_kernel<<<16384, tpb, 0, stream>>>(B2, 0.0f, (long)N * C3D);
  gat_agg_kernel<<<cdiv(total, 8), tpb, 0, stream>>>(ei, B1, a_s, a_d, emax, den, B2, E, total);
  gat_post_kernel<<<cdiv((long)N * C3D, 256), tpb, 0, stream>>>(B2, bg, B3, N);

  // --- global mean pool + classifier ---
  fill_kernel<<<cdiv(NGRAPH * (C3D + 1), 256) + 1, tpb, 0, stream>>>(
      pooled, 0.0f, (long)NGRAPH * C3D + NGRAPH);
  pool_kernel<<<cdiv(N, 8), tpb, 0, stream>>>(B3, batch, pooled, cnt, N);
  classifier_kernel<<<1, 64, 0, stream>>>(pooled, cnt, Wc1, bc1, Wc2, bc2, out);
}